// QuanvolutionClassifier_65481071395956
// MI455X (gfx1250) — compile-verified
//
#include <hip/hip_runtime.h>
#include <hip/hip_bf16.h>
#include <math.h>

typedef __attribute__((ext_vector_type(2))) float v2f;
typedef __attribute__((ext_vector_type(4))) float v4f;
typedef __attribute__((ext_vector_type(8))) float v8f;

#define EPSBN 1e-5f

// ---------------------------------------------------------------------------
// K0: zero stat accumulators, compute quantum scalar S, build padded fc_w^T
// ---------------------------------------------------------------------------
__global__ void qc_init(const float* __restrict__ theta,
                        const float* __restrict__ fc_w,
                        float* __restrict__ scal,     // [0..15] stats, [16] S
                        float* __restrict__ wT) {     // [784][16], cols>=10 zero
  int tid = threadIdx.x;
  if (tid < 16) scal[tid] = 0.f;
  if (tid == 16) {
    float p = 1.f;
    for (int i = 0; i < 24; ++i) p *= cosf(theta[i] * 0.5f);
    scal[16] = p * p;                 // probs[0]; probs[1..3] == 0
  }
  for (int i = tid; i < 784 * 16; i += blockDim.x) {
    int k = i >> 4, n = i & 15;
    wT[i] = (n < 10) ? fc_w[n * 784 + k] : 0.f;
  }
}

// ---------------------------------------------------------------------------
// K1..K4: out = conv3x3( f(in) ), 4 contiguous pixels per thread (b128 I/O).
//   MODE 0: f(u) = u
//   MODE 1: f(u) = relu(g*(u-mean)*rstd + b)
//   MODE 2: f(u) = relu(g*(u-mean)*rstd + b + x2)        (residual add)
// Zero 'SAME' padding applies to f(in), so invalid taps contribute 0.
// Emits per-block (sum, sumsq) partials of the conv output.
// ---------------------------------------------------------------------------
template <int MODE>
__global__ void __launch_bounds__(256)
qc_conv(const float* __restrict__ in,
        const float* __restrict__ addx,
        const float* __restrict__ w9,
        const float* __restrict__ gamma,
        const float* __restrict__ beta,
        const float* __restrict__ stats,   // [sum, sumsq] of `in`
        float* __restrict__ out,
        float* __restrict__ partials,
        int total4) {                      // B*784/4
  int idx = blockIdx.x * blockDim.x + threadIdx.x;

  float mean = 0.f, rstd = 0.f, g = 0.f, bb = 0.f;
  if (MODE != 0) {
    float n = (float)total4 * 4.f;
    float s0 = stats[0], s1 = stats[1];
    mean = s0 / n;
    float var = s1 / n - mean * mean;   // biased variance, as in reference
    rstd = rsqrtf(var + EPSBN);
    g = gamma[0];
    bb = beta[0];
  }
  float w[9];
#pragma unroll
  for (int i = 0; i < 9; ++i) w[i] = w9[i];

  float acc[4] = {0.f, 0.f, 0.f, 0.f};
  if (idx < total4) {
    int b  = idx / 196;                 // 196 quads per 28x28 plane
    int rm = idx - b * 196;
    int h  = rm / 7;
    int w0 = (rm - h * 7) * 4;          // 0,4,...,24
    const float* base  = in + (long)b * 784;
    const float* base2 = (MODE == 2) ? (addx + (long)b * 784) : nullptr;

    float win[3][6];                    // transformed cols w0-1 .. w0+4
#pragma unroll
    for (int dh = 0; dh < 3; ++dh) {
      int hh = h + dh - 1;
      if (hh >= 0 && hh < 28) {
        const float* rp = base + hh * 28;
        v4f c = *(const v4f*)(rp + w0);
        float l = (w0 > 0)  ? rp[w0 - 1] : 0.f;
        float r = (w0 < 24) ? rp[w0 + 4] : 0.f;
        v4f c2 = {0.f, 0.f, 0.f, 0.f};
        float l2 = 0.f, r2 = 0.f;
        if (MODE == 2) {
          const float* rp2 = base2 + hh * 28;
          c2 = *(const v4f*)(rp2 + w0);
          l2 = (w0 > 0)  ? rp2[w0 - 1] : 0.f;
          r2 = (w0 < 24) ? rp2[w0 + 4] : 0.f;
        }
        auto f = [&](float u, float a2, bool valid) -> float {
          if (!valid) return 0.f;
          if (MODE == 0) return u;
          float t = g * (u - mean) * rstd + bb;
          if (MODE == 2) t += a2;
          return fmaxf(t, 0.f);
        };
        win[dh][0] = f(l, l2, w0 > 0);
        win[dh][1] = f(c.x, c2.x, true);
        win[dh][2] = f(c.y, c2.y, true);
        win[dh][3] = f(c.z, c2.z, true);
        win[dh][4] = f(c.w, c2.w, true);
        win[dh][5] = f(r, r2, w0 < 24);
      } else {
#pragma unroll
        for (int k = 0; k < 6; ++k) win[dh][k] = 0.f;
      }
    }
#pragma unroll
    for (int j = 0; j < 4; ++j)
#pragma unroll
      for (int dh = 0; dh < 3; ++dh)
#pragma unroll
        for (int dw = 0; dw < 3; ++dw)
          acc[j] += win[dh][j + dw] * w[dh * 3 + dw];

    v4f ov = {acc[0], acc[1], acc[2], acc[3]};
    *(v4f*)(out + (long)idx * 4) = ov;  // idx*4 == b*784 + h*28 + w0
  }

  __shared__ float rs[256];
  __shared__ float rq[256];
  int tid = threadIdx.x;
  rs[tid] = acc[0] + acc[1] + acc[2] + acc[3];
  rq[tid] = acc[0] * acc[0] + acc[1] * acc[1] + acc[2] * acc[2] +
            acc[3] * acc[3];
  __syncthreads();
  for (int s = 128; s > 0; s >>= 1) {
    if (tid < s) { rs[tid] += rs[tid + s]; rq[tid] += rq[tid + s]; }
    __syncthreads();
  }
  if (tid == 0) {
    partials[2 * (long)blockIdx.x]     = rs[0];
    partials[2 * (long)blockIdx.x + 1] = rq[0];
  }
}

// ---------------------------------------------------------------------------
// Deterministic second-stage reduction of per-block partials -> dst[0..1]
// ---------------------------------------------------------------------------
__global__ void qc_reduce(const float* __restrict__ partials, int n,
                          float* __restrict__ dst) {
  __shared__ float rs[256];
  __shared__ float rq[256];
  float a = 0.f, b = 0.f;
  for (int i = threadIdx.x; i < n; i += 256) {
    a += partials[2 * (long)i];
    b += partials[2 * (long)i + 1];
  }
  int tid = threadIdx.x;
  rs[tid] = a;
  rq[tid] = b;
  __syncthreads();
  for (int s = 128; s > 0; s >>= 1) {
    if (tid < s) { rs[tid] += rs[tid + s]; rq[tid] += rq[tid + s]; }
    __syncthreads();
  }
  if (tid == 0) { dst[0] = rs[0]; dst[1] = rq[0]; }
}

// ---------------------------------------------------------------------------
// K5: fused = alpha*relu(bn4(t4)+y1) + (1-alpha)*q, y1 = relu(bn2(t2)+x),
// then logits = fused @ fc_w^T + fc_b via V_WMMA_F32_16X16X4_F32,
// then log_softmax.  One block = 16 batch rows, 4 waves split K=784.
// ---------------------------------------------------------------------------
__global__ void __launch_bounds__(128)
qc_fuse_fc(const float* __restrict__ x,
           const float* __restrict__ t2,
           const float* __restrict__ t4,
           const float* __restrict__ g2p, const float* __restrict__ b2p,
           const float* __restrict__ g4p, const float* __restrict__ b4p,
           const float* __restrict__ scal,   // stats + S
           const float* __restrict__ alphap,
           const float* __restrict__ wT,     // [784][16] padded transpose
           const float* __restrict__ fc_b,
           float* __restrict__ out, int B) {
  __shared__ float fused[16 * 788];   // 16 batch rows x 784 feats (pitch 788)
  __shared__ float red[4 * 256];      // per-wave 16x16 partial tiles
  __shared__ float logits[256];

  const float n = (float)B * 784.f;
  float mean2 = scal[2] / n;
  float rstd2 = rsqrtf(scal[3] / n - mean2 * mean2 + EPSBN);
  float mean4 = scal[6] / n;
  float rstd4 = rsqrtf(scal[7] / n - mean4 * mean4 + EPSBN);
  float gm2 = g2p[0], bt2 = b2p[0], gm4 = g4p[0], bt4 = b4p[0];
  float S = scal[16];
  float alpha = alphap[0];

  int tid = threadIdx.x;
  long m0 = (long)blockIdx.x * 16;

  // Phase 1: build fused features in LDS (b128 loads, b128 LDS stores)
  for (int i4 = tid; i4 < 16 * 196; i4 += 128) {
    int m  = i4 / 196;
    int j0 = (i4 - m * 196) * 4;              // multiple of 4
    v4f v = {0.f, 0.f, 0.f, 0.f};
    if (m0 + m < B) {
      long gi = (m0 + m) * 784 + j0;
      v4f a  = *(const v4f*)(t2 + gi);
      v4f c  = *(const v4f*)(t4 + gi);
      v4f xx = *(const v4f*)(x + gi);
#pragma unroll
      for (int e = 0; e < 4; ++e) {
        float y1 = fmaxf(gm2 * (a[e] - mean2) * rstd2 + bt2 + xx[e], 0.f);
        float y2 = fmaxf(gm4 * (c[e] - mean4) * rstd4 + bt4 + y1, 0.f);
        float q  = (e == 0) ? S : 0.f;        // j%4==0 pattern
        v[e] = alpha * y2 + (1.f - alpha) * q;
      }
    }
    *(v4f*)(&fused[m * 788 + j0]) = v;
  }
  __syncthreads();

  // Phase 2: WMMA f32 16x16x4 over this wave's K-chunk (196 K values -> 49 ops)
  int lane = tid & 31;
  int wave = tid >> 5;
  int half = lane >> 4;   // selects K pair within the 4-wide K step
  int lm   = lane & 15;   // M for A-frag, N for B-frag
  v8f cacc;
#pragma unroll
  for (int r = 0; r < 8; ++r) cacc[r] = 0.f;
  int kbase = wave * 196;
  for (int s = 0; s < 49; ++s) {
    int k = kbase + s * 4 + half * 2;
    v2f a, bf;
    a.x  = fused[lm * 788 + k];       // A[m=lm][k], A[m][k+1]
    a.y  = fused[lm * 788 + k + 1];
    bf.x = wT[(k) * 16 + lm];         // B[k][n=lm], B[k+1][n]
    bf.y = wT[(k + 1) * 16 + lm];
    cacc = __builtin_amdgcn_wmma_f32_16x16x4_f32(
        /*neg_a=*/false, a, /*neg_b=*/false, bf,
        /*c_mod=*/(short)0, cacc, /*reuse_a=*/false, /*reuse_b=*/false);
  }
#pragma unroll
  for (int r = 0; r < 8; ++r) {
    int m = r + half * 8;             // C/D layout: VGPR r -> M=r (+8 hi half)
    red[wave * 256 + m * 16 + lm] = cacc[r];
  }
  __syncthreads();

  // Phase 3: reduce 4 wave tiles, add bias
  for (int i = tid; i < 256; i += 128) {
    float v = red[i] + red[256 + i] + red[512 + i] + red[768 + i];
    int nn = i & 15;
    if (nn < 10) v += fc_b[nn];
    logits[i] = v;
  }
  __syncthreads();

  // Phase 4: log_softmax over the 10 valid classes, one thread per row
  if (tid < 16) {
    long row = m0 + tid;
    if (row < B) {
      float mx = -3.4e38f;
      for (int nn = 0; nn < 10; ++nn) mx = fmaxf(mx, logits[tid * 16 + nn]);
      float se = 0.f;
      for (int nn = 0; nn < 10; ++nn) se += expf(logits[tid * 16 + nn] - mx);
      float lse = mx + logf(se);
      for (int nn = 0; nn < 10; ++nn)
        out[row * 10 + nn] = logits[tid * 16 + nn] - lse;
    }
  }
}

// ---------------------------------------------------------------------------
extern "C" void kernel_launch(void* const* d_in, const int* in_sizes, int n_in,
                              void* d_out, int out_size, void* d_ws,
                              size_t ws_size, hipStream_t stream) {
  const float* x     = (const float*)d_in[0];
  const float* theta = (const float*)d_in[1];
  const float* r1_w1 = (const float*)d_in[2];
  const float* r1_g1 = (const float*)d_in[3];
  const float* r1_b1 = (const float*)d_in[4];
  const float* r1_w2 = (const float*)d_in[5];
  const float* r1_g2 = (const float*)d_in[6];
  const float* r1_b2 = (const float*)d_in[7];
  const float* r2_w1 = (const float*)d_in[8];
  const float* r2_g1 = (const float*)d_in[9];
  const float* r2_b1 = (const float*)d_in[10];
  const float* r2_w2 = (const float*)d_in[11];
  const float* r2_g2 = (const float*)d_in[12];
  const float* r2_b2 = (const float*)d_in[13];
  const float* fc_w  = (const float*)d_in[14];
  const float* fc_b  = (const float*)d_in[15];
  const float* alpha = (const float*)d_in[16];
  float* out = (float*)d_out;
  float* wsf = (float*)d_ws;

  int  B      = in_sizes[0] / 784;
  long NE     = (long)B * 784;
  int  total4 = (int)(NE / 4);
  int  nblk   = (total4 + 255) / 256;

  // ws layout (floats): [0..16] scalars | [1024..) wT | [16384..) partials |
  // [131072..) three ping-pong planes of NE floats each.
  float* scal     = wsf;
  float* wT       = wsf + 1024;
  float* partials = wsf + 16384;
  float* bufA     = wsf + 131072;     // t1, later t3
  float* bufB     = bufA + NE;        // t2 (kept for residual in K5)
  float* bufC     = bufB + NE;        // t4

  qc_init<<<1, 256, 0, stream>>>(theta, fc_w, scal, wT);

  // t1 = conv1(x)
  qc_conv<0><<<nblk, 256, 0, stream>>>(x, nullptr, r1_w1, r1_g1, r1_b1, scal,
                                       bufA, partials, total4);
  qc_reduce<<<1, 256, 0, stream>>>(partials, nblk, scal + 0);
  // t2 = conv2(relu(bn1(t1)))
  qc_conv<1><<<nblk, 256, 0, stream>>>(bufA, nullptr, r1_w2, r1_g1, r1_b1,
                                       scal + 0, bufB, partials, total4);
  qc_reduce<<<1, 256, 0, stream>>>(partials, nblk, scal + 2);
  // t3 = conv3(relu(bn2(t2) + x))          (y1 computed inline on taps)
  qc_conv<2><<<nblk, 256, 0, stream>>>(bufB, x, r2_w1, r1_g2, r1_b2,
                                       scal + 2, bufA, partials, total4);
  qc_reduce<<<1, 256, 0, stream>>>(partials, nblk, scal + 4);
  // t4 = conv4(relu(bn3(t3)))
  qc_conv<1><<<nblk, 256, 0, stream>>>(bufA, nullptr, r2_w2, r2_g1, r2_b1,
                                       scal + 4, bufC, partials, total4);
  qc_reduce<<<1, 256, 0, stream>>>(partials, nblk, scal + 6);

  // fused features + FC (WMMA) + log_softmax
  int nb5 = (B + 15) / 16;
  qc_fuse_fc<<<nb5, 128, 0, stream>>>(x, bufB, bufC, r1_g2, r1_b2, r2_g2,
                                      r2_b2, scal, alpha, wT, fc_b, out, B);
}